// MultiPosCrossEntropyLoss_63883343560926
// MI455X (gfx1250) — compile-verified
//
#include <hip/hip_runtime.h>
#include <hip/hip_bf16.h>
#include <stdint.h>

// MultiPosCrossEntropyLoss for MI455X (gfx1250).
// Bandwidth-bound: 201 MB read -> ~8.6 us floor at 23.3 TB/s. No matmul
// structure, so no WMMA; instead use the CDNA5 async global->LDS path
// (ASYNCcnt) for VGPR-free double-buffered streaming, wave32 shuffle
// reductions, and one f32 atomic per row.

namespace {
constexpr int kN = 8192;            // rows
constexpr int kC = 2048;            // cols
constexpr int kThreads = 256;       // 8 waves (wave32)
constexpr int kRowsPerBlock = 4;    // grid = 2048 blocks
constexpr int kWaves = kThreads / 32;
constexpr float kInvN = 1.0f / float(kN);
}

// CDNA5 async copy: global memory -> LDS, 16B per lane, tracked by ASYNCcnt.
// GVS form: SGPR64 base + VGPR32 byte offset; first operand is LDS byte addr.
__device__ __forceinline__ void async_copy_b128(unsigned lds_addr, unsigned goff,
                                                const void* base) {
  asm volatile("global_load_async_to_lds_b128 %0, %1, %2"
               :: "v"(lds_addr), "v"(goff), "s"(base)
               : "memory");
}

__global__ __launch_bounds__(kThreads)
void mpce_loss_kernel(const float* __restrict__ score,
                      const void* __restrict__ label,   // int64 labels, 0/1
                      float* __restrict__ out) {
  __shared__ float s_score[2][kC];                 // 2 x 8 KB
  __shared__ unsigned long long s_label[2][kC];    // 2 x 16 KB
  __shared__ float s_red_n[kWaves];
  __shared__ float s_red_p[kWaves];

  const int t    = threadIdx.x;
  const int lane = t & 31;
  const int wave = t >> 5;
  const int row0 = blockIdx.x * kRowsPerBlock;

  // Issue one row's worth of async b128 copies (2 score + 4 label per thread).
  auto issue = [&](int buf, int row) {
    unsigned sb = (unsigned)(uintptr_t)&s_score[buf][0];  // LDS byte offset
    unsigned lb = (unsigned)(uintptr_t)&s_label[buf][0];
    unsigned gs = (unsigned)row * (unsigned)(kC * 4);
    unsigned gl = (unsigned)row * (unsigned)(kC * 8);
#pragma unroll
    for (int i = 0; i < 2; ++i) {
      unsigned c16 = ((unsigned)t + (unsigned)(i * kThreads)) * 16u;
      async_copy_b128(sb + c16, gs + c16, (const void*)score);
    }
#pragma unroll
    for (int i = 0; i < 4; ++i) {
      unsigned c16 = ((unsigned)t + (unsigned)(i * kThreads)) * 16u;
      async_copy_b128(lb + c16, gl + c16, label);
    }
  };

  issue(0, row0);  // prologue: fill buffer 0

  for (int r = 0; r < kRowsPerBlock; ++r) {
    if (r + 1 < kRowsPerBlock) {
      issue((r + 1) & 1, row0 + r + 1);            // prefetch next row
      // 6 newest ops (next row) may be outstanding; older (this row) must be
      // done. Async loads complete in order, so <=6 means buffer r is ready.
      asm volatile("s_wait_asynccnt 6" ::: "memory");
    } else {
      asm volatile("s_wait_asynccnt 0" ::: "memory");
    }
    __syncthreads();  // every wave waited for its own slice -> buffer valid

    const float*    ss = &s_score[r & 1][0];
    const unsigned* lw = (const unsigned*)&s_label[r & 1][0];  // low dwords
    float sn = 0.f, sp = 0.f;
#pragma unroll
    for (int k = 0; k < kC / kThreads; ++k) {      // strided: no bank conflicts
      int e = t + k * kThreads;
      float x = ss[e];
      unsigned lab = lw[2 * e];                    // int64 low word (0 or 1)
      float v  = (lab == 0u) ? x : -x;
      float ex = __expf(v);
      sn += (lab == 0u) ? ex : 0.f;
      sp += (lab == 1u) ? ex : 0.f;
    }

    // wave32 tree reduction
#pragma unroll
    for (int off = 16; off > 0; off >>= 1) {
      sn += __shfl_down(sn, off, 32);
      sp += __shfl_down(sp, off, 32);
    }
    if (lane == 0) { s_red_n[wave] = sn; s_red_p[wave] = sp; }
    __syncthreads();
    if (t == 0) {
      float tn = 0.f, tp = 0.f;
#pragma unroll
      for (int w = 0; w < kWaves; ++w) { tn += s_red_n[w]; tp += s_red_p[w]; }
      // softplus(log(sum_neg) + log(sum_pos)) == log1p(sum_neg * sum_pos);
      // empty pos/neg set -> 0 product -> loss 0, matching the reference.
      atomicAdd(out, log1pf(tn * tp) * kInvN);
    }
    __syncthreads();  // protect red[] and buffer (r&1) before next overwrite
  }
}

__global__ void zero_out_kernel(float* out) { out[0] = 0.0f; }

extern "C" void kernel_launch(void* const* d_in, const int* in_sizes, int n_in,
                              void* d_out, int out_size, void* d_ws, size_t ws_size,
                              hipStream_t stream) {
  const float* score = (const float*)d_in[0];
  const void*  label = d_in[1];            // int64 [N, C]
  float* out = (float*)d_out;

  zero_out_kernel<<<1, 1, 0, stream>>>(out);
  mpce_loss_kernel<<<kN / kRowsPerBlock, kThreads, 0, stream>>>(score, label, out);

  (void)in_sizes; (void)n_in; (void)out_size; (void)d_ws; (void)ws_size;
}